// MultiLossFunc_40037685133315
// MI455X (gfx1250) — compile-verified
//
#include <hip/hip_runtime.h>
#include <math.h>

// ---------------- problem constants ----------------
#define CN    100
#define KPAT  4
#define NC    400          // CN*KPAT
#define DIMN  512
#define BATCH 2048
#define TWOB  4096
#define TAU_  0.1f
#define T1_   0.1f
#define T2_   0.1f
#define THR_  0.95f

// con-loss block tiling
#define BM 64
#define BN 64
#define BK 32

typedef __attribute__((ext_vector_type(2))) float v2f;
typedef __attribute__((ext_vector_type(8))) float v8f;
typedef __attribute__((ext_vector_type(4))) unsigned int u32x4;
typedef __attribute__((ext_vector_type(4))) int i32x4;
typedef __attribute__((ext_vector_type(8))) int i32x8;

#if __has_builtin(__builtin_amdgcn_tensor_load_to_lds) && __has_builtin(__builtin_amdgcn_s_wait_tensorcnt)
#define HAVE_TDM 1
#else
#define HAVE_TDM 0
#endif

// ---------------- workspace layout (float indices) ----------------
#define OFF_CENTERS 0          // 512*400 = 204800
#define OFF_SIM     204800     // 2048*400 = 819200
#define OFF_CLS     1024000    // 400  class pattern sums
#define OFF_CNT     1024400    // 100 (+pad 12)
#define OFF_ROWE    1024512    // 4096 row sum exp(logits)
#define OFF_ROWN    1028608    // 4096 row sum mask*w*logit
#define OFF_ROWM    1032704    // 4096 row mask count
#define OFF_SCAL    1036800    // 16 scalars: [0]=classify acc [1]=regdiff acc
#define ZERO_BASE   1024000
#define ZERO_N      12816
#define OFF_PL      1036816    // 2048 ints (pseudo labels)

// ---------------- helpers ----------------
__device__ __forceinline__ float wave_sum32(float v) {
#pragma unroll
  for (int m = 16; m; m >>= 1) v += __shfl_xor(v, m, 32);
  return v;
}
// reduce independently within each 16-lane half of the wave
__device__ __forceinline__ float half_sum16(float v) {
#pragma unroll
  for (int m = 8; m; m >>= 1) v += __shfl_xor(v, m, 32);
  return v;
}

__device__ __forceinline__ v8f wmma_f32_16x16x4(v2f a, v2f b, v8f c) {
  return __builtin_amdgcn_wmma_f32_16x16x4_f32(false, a, false, b, (short)0, c,
                                               false, false);
}

#if HAVE_TDM
// Issue one TDM 2D tile load: 64 rows x 32 f32, row stride 512 f32, compact in LDS.
// D# bit layout per CDNA5 ISA ch.8 (group0: count/lds/global/type, group1: dims).
__device__ __forceinline__ void tdm_load_tile(unsigned long long ga, unsigned lds_off) {
  u32x4 g0;
  g0[0] = 1u;                                                  // count=1, user desc
  g0[1] = lds_off;                                             // lds_addr [63:32]
  g0[2] = (unsigned)(ga & 0xFFFFFFFFull);                      // global_addr[31:0]
  g0[3] = ((unsigned)(ga >> 32) & 0x01FFFFFFu) | (2u << 30);   // ga[56:32] | type=2
  i32x8 g1;
  g1[0] = 0x00020000;            // data_size=2 (4B), no multicast/pad/iterate
  g1[1] = (int)(BK << 16);       // tensor_dim0 = 32   (bits 79:48 low half)
  g1[2] = (int)(BM << 16);       // tensor_dim1 = 64   (bits 111:80 low half)
  g1[3] = (int)(BK << 16);       // tile_dim0 = 32     (bits 127:112)
  g1[4] = BM;                    // tile_dim1 = 64, tile_dim2 = 0
  g1[5] = DIMN;                  // tensor_dim0_stride = 512
  g1[6] = 0;
  g1[7] = 0;                     // tensor_dim1_stride unused
  i32x4 z4 = {0, 0, 0, 0};
#if defined(__clang_major__) && (__clang_major__ >= 23)
  i32x8 z8 = {0, 0, 0, 0, 0, 0, 0, 0};
  __builtin_amdgcn_tensor_load_to_lds(g0, g1, z4, z4, z8, 0);
#else
  __builtin_amdgcn_tensor_load_to_lds(g0, g1, z4, z4, 0);
#endif
}
#endif

// ---------------- kernels ----------------
__global__ void k_zero(float* z, int n) {
  int i = blockIdx.x * blockDim.x + threadIdx.x;
  if (i < n) z[i] = 0.0f;
}

// centers = weight / ||col||   (weight is DIM x NC row-major)
__global__ void k_centers(const float* __restrict__ w, float* __restrict__ centers) {
  int c = blockIdx.x * blockDim.x + threadIdx.x;
  if (c >= NC) return;
  float s = 0.f;
  for (int d = 0; d < DIMN; ++d) { float v = w[(size_t)d * NC + c]; s += v * v; }
  float inv = rsqrtf(s);
  for (int d = 0; d < DIMN; ++d) centers[(size_t)d * NC + c] = w[(size_t)d * NC + c] * inv;
}

// sim = weak(2048x512) @ centers(512x400)  -- one 16x16 tile per wave, f32 WMMA
__global__ void k_sim_wmma(const float* __restrict__ A, const float* __restrict__ Bm,
                           float* __restrict__ sim) {
  int wid  = (blockIdx.x * blockDim.x + threadIdx.x) >> 5;   // 3200 waves exact
  int lane = threadIdx.x & 31;
  int tm = wid / (NC / 16);
  int tn = wid % (NC / 16);
  int m = tm * 16 + (lane & 15);
  int n = tn * 16 + (lane & 15);
  int ksel = (lane >> 4) << 1;
  const float2* pA = (const float2*)(A + (size_t)m * DIMN);
  v8f acc = {};
  for (int k0 = 0; k0 < DIMN; k0 += 4) {
    int kb = k0 + ksel;
    float2 av = pA[kb >> 1];
    v2f af = {av.x, av.y};
    v2f bf = {Bm[(size_t)kb * NC + n], Bm[(size_t)(kb + 1) * NC + n]};
    acc = wmma_f32_16x16x4(af, bf, acc);
  }
#pragma unroll
  for (int r = 0; r < 8; ++r) {
    int row = tm * 16 + r + ((lane & 16) ? 8 : 0);
    int col = tn * 16 + (lane & 15);
    sim[(size_t)row * NC + col] = acc[r];
  }
}

// per-row (one wave per sample): classify loss, pseudo labels, segment sums, predict
__global__ void k_rowloss(const float* __restrict__ sim, const int* __restrict__ labels,
                          float* classSums, float* counts, float* scal,
                          int* pl, float* outPredict) {
  int b    = (blockIdx.x * blockDim.x + threadIdx.x) >> 5;   // 2048 waves exact
  int lane = threadIdx.x & 31;
  const float* row = sim + (size_t)b * NC;

  float cmax[4];
  int   ncls = 0;
  float vmax = -3.4e38f; int vidx = 0x7fffffff;
  for (int c = lane; c < CN; c += 32) {
    float4 q = *(const float4*)(row + 4 * c);
    float m = fmaxf(fmaxf(q.x, q.y), fmaxf(q.z, q.w));
    cmax[ncls++] = m;
    if (m > vmax) { vmax = m; vidx = c; }
  }
  float gmax = vmax; int gidx = vidx;          // argmax (first occurrence)
#pragma unroll
  for (int m = 16; m; m >>= 1) {
    float ov = __shfl_xor(gmax, m, 32);
    int   oi = __shfl_xor(gidx, m, 32);
    if (ov > gmax || (ov == gmax && oi < gidx)) { gmax = ov; gidx = oi; }
  }
  float S = 0.f;
  for (int t = 0; t < ncls; ++t) S += expf((cmax[t] - gmax) / T1_);
  S = wave_sum32(S);

  if (lane == 0) {
    int L = labels[b];
    float4 q = *(const float4*)(row + 4 * L);
    float sk[4] = {q.x, q.y, q.z, q.w};
    float smL = fmaxf(fmaxf(sk[0], sk[1]), fmaxf(sk[2], sk[3]));
    float wk[4], wden = 0.f;
#pragma unroll
    for (int k = 0; k < 4; ++k) { wk[k] = expf((sk[k] - smL) / TAU_); wden += wk[k]; }
#pragma unroll
    for (int k = 0; k < 4; ++k) wk[k] /= wden;

    float eL = expf((smL - gmax) / T1_);
    float cls = 0.f;
#pragma unroll
    for (int k = 0; k < 4; ++k) {
      float ek   = expf((sk[k] - gmax) / T1_);
      float logp = (sk[k] - gmax) / T1_ - logf(S - eL + ek);
      cls += wk[k] * (-logp);
    }
    atomicAdd(scal + 0, cls);

    float mp = wk[0]; int pi = 0;
#pragma unroll
    for (int k = 1; k < 4; ++k) if (wk[k] > mp) { mp = wk[k]; pi = k; }
    pl[b] = (mp >= THR_) ? (L * KPAT + pi) : (b + CN * KPAT);

#pragma unroll
    for (int k = 0; k < 4; ++k) atomicAdd(classSums + L * KPAT + k, wk[k]);
    atomicAdd(counts + L, 1.0f);
    outPredict[b] = (float)gidx;
  }
}

// reg_loss_diff: 600 within-class pairs of center columns, one wave per pair
__global__ void k_regdiff(const float* __restrict__ centers, float* scal) {
  int p    = (blockIdx.x * blockDim.x + threadIdx.x) >> 5;
  int lane = threadIdx.x & 31;
  if (p >= CN * 6) return;
  static const int PA[6] = {0, 0, 0, 1, 1, 2};
  static const int PB[6] = {1, 2, 3, 2, 3, 3};
  int g = p / 6, q = p % 6;
  int ci = g * KPAT + PA[q], cj = g * KPAT + PB[q];
  float dot = 0.f;
  for (int d = lane; d < DIMN; d += 32)
    dot += centers[(size_t)d * NC + ci] * centers[(size_t)d * NC + cj];
  dot = wave_sum32(dot);
  if (lane == 0) atomicAdd(scal + 1, fabsf(dot));
}

// one K-chunk of the block tile: 8 WMMA per accumulator, A fragment shared
__device__ __forceinline__ void con_chunk(const float* A, const float* B,
                                          int mloc, int nloc0, int ksel,
                                          v8f& acc0, v8f& acc1) {
#pragma unroll
  for (int kk = 0; kk < BK; kk += 4) {
    int kb = kk + ksel;
    float2 av = *(const float2*)(A + mloc * BK + kb);
    float2 b0 = *(const float2*)(B + nloc0 * BK + kb);
    float2 b1 = *(const float2*)(B + (nloc0 + 16) * BK + kb);
    v2f af  = {av.x, av.y};
    v2f bf0 = {b0.x, b0.y};
    v2f bf1 = {b1.x, b1.y};
    acc0 = wmma_f32_16x16x4(af, bf0, acc0);
    acc1 = wmma_f32_16x16x4(af, bf1, acc1);
  }
}

// con-loss: 64x64 sim block tile per workgroup, TDM double-buffered LDS staging,
// streamed row statistics (no 64MB sim matrix in HBM).
__global__ void k_con_wmma(const float* __restrict__ weak, const float* __restrict__ strong,
                           const int* __restrict__ pl, const int* __restrict__ cur_epoch,
                           float* rowE, float* rowN, float* rowM) {
  __shared__ float lA[2][BM * BK];
  __shared__ float lB[2][BN * BK];
  int tid  = threadIdx.x;
  int wave = tid >> 5, lane = tid & 31;
  int bm = blockIdx.x >> 6, bn = blockIdx.x & 63;
  int row0 = bm * BM, col0 = bn * BN;
  const float* Abase = (row0 < BATCH) ? weak + (size_t)row0 * DIMN
                                      : strong + (size_t)(row0 - BATCH) * DIMN;
  const float* Bbase = (col0 < BATCH) ? weak + (size_t)col0 * DIMN
                                      : strong + (size_t)(col0 - BATCH) * DIMN;

  int msub  = wave >> 1;
  int nsub0 = (wave & 1) * 2;
  int mloc  = msub * 16 + (lane & 15);
  int nloc0 = nsub0 * 16 + (lane & 15);
  int ksel  = (lane >> 4) << 1;
  v8f acc0 = {}, acc1 = {};

#if HAVE_TDM
  if (wave == 0) {
    tdm_load_tile((unsigned long long)(uintptr_t)Abase, (unsigned)(uintptr_t)&lA[0][0]);
    tdm_load_tile((unsigned long long)(uintptr_t)Bbase, (unsigned)(uintptr_t)&lB[0][0]);
  }
  for (int c = 0; c < DIMN / BK; ++c) {
    int buf = c & 1;
    if (wave == 0) {
      if (c + 1 < DIMN / BK) {
        tdm_load_tile((unsigned long long)(uintptr_t)(Abase + (c + 1) * BK),
                      (unsigned)(uintptr_t)&lA[buf ^ 1][0]);
        tdm_load_tile((unsigned long long)(uintptr_t)(Bbase + (c + 1) * BK),
                      (unsigned)(uintptr_t)&lB[buf ^ 1][0]);
        __builtin_amdgcn_s_wait_tensorcnt(2);   // chunk c done (in-order TDM)
      } else {
        __builtin_amdgcn_s_wait_tensorcnt(0);
      }
    }
    __syncthreads();
    con_chunk(&lA[buf][0], &lB[buf][0], mloc, nloc0, ksel, acc0, acc1);
    __syncthreads();
  }
#else
  for (int c = 0; c < DIMN / BK; ++c) {
    __syncthreads();
    for (int idx = tid; idx < BM * BK / 4; idx += 256) {
      int r = idx >> 3, col = (idx & 7) * 4;
      ((float4*)&lA[0][0])[idx] = *(const float4*)(Abase + (size_t)r * DIMN + c * BK + col);
      ((float4*)&lB[0][0])[idx] = *(const float4*)(Bbase + (size_t)r * DIMN + c * BK + col);
    }
    __syncthreads();
    con_chunk(&lA[0][0], &lB[0][0], mloc, nloc0, ksel, acc0, acc1);
  }
#endif

  bool use_pl = (*cur_epoch) >= 5;
#pragma unroll
  for (int t = 0; t < 2; ++t) {
    v8f acc = t ? acc1 : acc0;
    int j   = col0 + nsub0 * 16 + t * 16 + (lane & 15);
    int plj = pl[j & (BATCH - 1)];
#pragma unroll
    for (int r = 0; r < 8; ++r) {
      int i = row0 + msub * 16 + r + ((lane & 16) ? 8 : 0);
      float s = acc[r];
      bool diag = (i == j);
      bool same = ((i & (BATCH - 1)) == (j & (BATCH - 1)));
      int  pli  = pl[i & (BATCH - 1)];
      bool msk  = !diag && (use_pl ? (pli == plj) : same);
      float wgt = use_pl ? (same ? 1.0f : s) : 1.0f;
      float e   = diag ? 0.0f : expf(s / T2_);
      float num = msk ? wgt * (s / T2_) : 0.0f;
      float cnt = msk ? 1.0f : 0.0f;
      e = half_sum16(e); num = half_sum16(num); cnt = half_sum16(cnt);
      if ((lane & 15) == 0) {
        atomicAdd(rowE + i, e);
        atomicAdd(rowN + i, num);
        atomicAdd(rowM + i, cnt);
      }
    }
  }
}

// finalize: reg_loss (class entropy * variance scale), con loss mean, scalar outs
__global__ void k_finalize(const float* __restrict__ cv, const int* __restrict__ cur_epoch,
                           const float* __restrict__ classSums, const float* __restrict__ counts,
                           const float* __restrict__ rowE, const float* __restrict__ rowN,
                           const float* __restrict__ rowM, const float* __restrict__ scal,
                           float* out) {
  __shared__ float ra[256];
  __shared__ float rb[256];
  int t = threadIdx.x;
  float mn = (t < CN) ? cv[t] : 3.4e38f;
  float mx = (t < CN) ? cv[t] : -3.4e38f;
  ra[t] = mn; rb[t] = mx; __syncthreads();
  for (int s = 128; s > 0; s >>= 1) {
    if (t < s) { ra[t] = fminf(ra[t], ra[t + s]); rb[t] = fmaxf(rb[t], rb[t + s]); }
    __syncthreads();
  }
  float cmn = ra[0], cmx = rb[0];
  __syncthreads();
  int ep = *cur_epoch;
  float contrib = 0.f, pres = 0.f;
  if (t < CN) {
    float cnt = counts[t];
    if (cnt > 0.f) {
      float denom = fmaxf(cnt, 1.0f);
      float ent = logf((float)KPAT);
#pragma unroll
      for (int k = 0; k < KPAT; ++k) {
        float ave = 0.9f * (classSums[t * KPAT + k] / denom) + 0.1f / KPAT;
        ent += ave * logf(ave);
      }
      float scale;
      if (ep == 0) scale = cv[t];
      else { float var = (cv[t] - cmn) / (cmx - cmn); scale = expf((var - 1.0f) * 1.5f); }
      contrib = ent * scale;
      pres = 1.0f;
    }
  }
  ra[t] = contrib; rb[t] = pres; __syncthreads();
  for (int s = 128; s > 0; s >>= 1) {
    if (t < s) { ra[t] += ra[t + s]; rb[t] += rb[t + s]; }
    __syncthreads();
  }
  float reg = ra[0] / fmaxf(rb[0], 1.0f);
  __syncthreads();
  float acc = 0.f;
  for (int i = t; i < TWOB; i += 256) acc += rowN[i] / rowM[i] - logf(rowE[i]);
  ra[t] = acc; __syncthreads();
  for (int s = 128; s > 0; s >>= 1) { if (t < s) ra[t] += ra[t + s]; __syncthreads(); }
  if (t == 0) {
    out[0] = scal[0] / (float)BATCH;       // classify_loss
    out[1] = -ra[0] / (float)TWOB;         // sup_con_loss
    out[2] = reg;                          // reg_loss
    out[3] = scal[1] / 600.0f;             // reg_loss_diff
  }
}

// ---------------- launch ----------------
extern "C" void kernel_launch(void* const* d_in, const int* in_sizes, int n_in,
                              void* d_out, int out_size, void* d_ws, size_t ws_size,
                              hipStream_t stream) {
  const float* weak      = (const float*)d_in[0];
  const float* strong    = (const float*)d_in[1];
  const float* weight    = (const float*)d_in[2];
  const float* class_var = (const float*)d_in[3];
  const int*   labels    = (const int*)d_in[4];
  const int*   cur_epoch = (const int*)d_in[5];
  float* out = (float*)d_out;
  float* wsf = (float*)d_ws;

  float* centers   = wsf + OFF_CENTERS;
  float* sim       = wsf + OFF_SIM;
  float* classSums = wsf + OFF_CLS;
  float* counts    = wsf + OFF_CNT;
  float* rowE      = wsf + OFF_ROWE;
  float* rowN      = wsf + OFF_ROWN;
  float* rowM      = wsf + OFF_ROWM;
  float* scal      = wsf + OFF_SCAL;
  int*   pl        = (int*)(wsf + OFF_PL);

  k_zero<<<(ZERO_N + 255) / 256, 256, 0, stream>>>(wsf + ZERO_BASE, ZERO_N);
  k_centers<<<(NC + 255) / 256, 256, 0, stream>>>(weight, centers);
  k_sim_wmma<<<800, 128, 0, stream>>>(weak, centers, sim);
  k_rowloss<<<256, 256, 0, stream>>>(sim, labels, classSums, counts, scal, pl, out + 4);
  k_regdiff<<<(600 * 32 + 127) / 128, 128, 0, stream>>>(centers, scal);
  k_con_wmma<<<4096, 256, 0, stream>>>(weak, strong, pl, cur_epoch, rowE, rowN, rowM);
  k_finalize<<<1, 256, 0, stream>>>(class_var, cur_epoch, classSums, counts,
                                    rowE, rowN, rowM, scal, out);
}